// CDOD_9423158247656
// MI455X (gfx1250) — compile-verified
//
#include <hip/hip_runtime.h>
#include <math.h>

#define NN 2048      // nodes
#define DF 128       // node feature / memory dim
#define DTF 8        // time features
#define KNB 20       // neighbors
#define HD 512       // hidden dim
#define EE 131072    // events
#define DIVB 8192    // chunk size
#define NCHUNK (EE / DIVB)

typedef float v2f __attribute__((ext_vector_type(2)));
typedef float v8f __attribute__((ext_vector_type(8)));

#if defined(__gfx1250__) && __has_builtin(__builtin_amdgcn_wmma_f32_16x16x4_f32)
#define HAVE_WMMA_F32 1
#endif

// D(16x16,f32) += A(16x4,f32) * B(4x16,f32), wave32.
// A layout: lane L<16 holds A[L, k0], A[L, k0+1]; lane L+16 holds A[L, k0+2],
// A[L, k0+3]. B mirrors A with N in place of M. C: VGPR i = C[i+8*(lane>=16),
// lane&15].
__device__ __forceinline__ v8f wmma4(v2f a, v2f b, v8f c) {
#ifdef HAVE_WMMA_F32
  return __builtin_amdgcn_wmma_f32_16x16x4_f32(false, a, false, b, (short)0, c,
                                               false, false);
#else
  int lane = threadIdx.x & 31;
  int row8 = (lane >> 4) << 3;
  int col = lane & 15;
#pragma unroll
  for (int i = 0; i < 8; ++i) {
    int row = i + row8;
    float s = 0.f;
    s += __shfl(a.x, row) * __shfl(b.x, col);
    s += __shfl(a.y, row) * __shfl(b.y, col);
    s += __shfl(a.x, row + 16) * __shfl(b.x, col + 16);
    s += __shfl(a.y, row + 16) * __shfl(b.y, col + 16);
    c[i] += s;
  }
  return c;
#endif
}

__device__ __forceinline__ float sigm(float x) { return 1.f / (1.f + expf(-x)); }

// ---------------------------------------------------------------------------
// Generic GEMM: C[M,Nt] = A[M,K] @ B (+bias)(+relu).
// BT=true: B stored [Nt,K] (x @ W.T, W row-major; ldb must be EVEN).
// BT=false: B stored [K,Nt].
// One wave computes a 32x64 macro tile (8 WMMA accumulators).
// Requirements: M%32==0, Nt%64==0, K%4==0, lda even.
// ---------------------------------------------------------------------------
template <bool BT>
__global__ void k_gemm(const float* __restrict__ A, int lda,
                       const float* __restrict__ B, int ldb,
                       const float* __restrict__ bias, float* __restrict__ C,
                       int ldc, int M, int Nt, int K, int relu) {
  int wid = blockIdx.x * 8 + (threadIdx.x >> 5);
  int ntiles = Nt >> 6;
  int total = (M >> 5) * ntiles;
  if (wid >= total) return;  // wave-uniform: EXEC stays all-ones for WMMA
  int lane = threadIdx.x & 31;
  int r = lane & 15, half = lane >> 4, kh2 = half << 1;
  int tm = (wid / ntiles) << 5;
  int tn = (wid % ntiles) << 6;
  const float* arow0 = A + (size_t)(tm + r) * lda;
  const float* arow1 = arow0 + (size_t)16 * lda;
  v8f c00 = {}, c01 = {}, c02 = {}, c03 = {};
  v8f c10 = {}, c11 = {}, c12 = {}, c13 = {};
  for (int k0 = 0; k0 < K; k0 += 4) {
    int ka = k0 + kh2;
    v2f a0 = *(const v2f*)(arow0 + ka);
    v2f a1 = *(const v2f*)(arow1 + ka);
#define KG_STEP(T, CA, CB)                              \
  {                                                     \
    int col = tn + (T << 4) + r;                        \
    v2f b;                                              \
    if (BT) {                                           \
      b = *(const v2f*)(B + (size_t)col * ldb + ka);    \
    } else {                                            \
      b.x = B[(size_t)ka * ldb + col];                  \
      b.y = B[(size_t)(ka + 1) * ldb + col];            \
    }                                                   \
    CA = wmma4(a0, b, CA);                              \
    CB = wmma4(a1, b, CB);                              \
  }
    KG_STEP(0, c00, c10)
    KG_STEP(1, c01, c11)
    KG_STEP(2, c02, c12)
    KG_STEP(3, c03, c13)
#undef KG_STEP
  }
#define KG_STORE(T, CA, CB)                                       \
  {                                                               \
    int col = tn + (T << 4) + r;                                  \
    float bv = bias ? bias[col] : 0.f;                            \
    for (int i = 0; i < 8; ++i) {                                 \
      float v0 = CA[i] + bv;                                      \
      float v1 = CB[i] + bv;                                      \
      if (relu) {                                                 \
        v0 = fmaxf(v0, 0.f);                                      \
        v1 = fmaxf(v1, 0.f);                                      \
      }                                                           \
      C[(size_t)(tm + i + (half << 3)) * ldc + col] = v0;         \
      C[(size_t)(tm + 16 + i + (half << 3)) * ldc + col] = v1;    \
    }                                                             \
  }
  KG_STORE(0, c00, c10)
  KG_STORE(1, c01, c11)
  KG_STORE(2, c02, c12)
  KG_STORE(3, c03, c13)
#undef KG_STORE
}

// ---------------------------------------------------------------------------
// Memory-GRU input GEMM: gi = msg[idx] @ Wi_m.T with msg gathered on the fly
// msg = [mem[d] (0..127) | edge_feat (128) | node_feat[d] (129..256)], K=257
// Segmented, branch-free A fetch; Wi is [384,257] row-major (odd stride).
// ---------------------------------------------------------------------------
__global__ void k_gemm_gi_mem(const float* __restrict__ mem,
                              const float* __restrict__ nf,
                              const int* __restrict__ lastpos,
                              const int* __restrict__ dstC,
                              const float* __restrict__ efC,
                              const float* __restrict__ Wi,
                              float* __restrict__ gi) {
  const int Nt = 384, LB = 257;
  int wid = blockIdx.x * 8 + (threadIdx.x >> 5);
  int ntiles = Nt >> 6;  // 6
  if (wid >= (NN >> 4) * ntiles) return;
  int lane = threadIdx.x & 31;
  int r = lane & 15, half = lane >> 4, kh2 = half << 1;
  int tm = (wid / ntiles) << 4;
  int tn = (wid % ntiles) << 6;
  int n = tm + r;
  int lp = lastpos[n];
  int e = lp < 0 ? 0 : lp;
  int d = dstC[e];
  float efv = efC[e];
  const float* mrow = mem + (size_t)d * DF;
  const float* frow = nf + (size_t)d * DF;
  v8f c0 = {}, c1 = {}, c2 = {}, c3 = {};
#define GI_MM(AV, KA)                                  \
  {                                                    \
    v2f bb;                                            \
    int col = tn + r;                                  \
    bb.x = Wi[(size_t)col * LB + (KA)];                \
    bb.y = Wi[(size_t)col * LB + (KA) + 1];            \
    c0 = wmma4(AV, bb, c0);                            \
    col = tn + 16 + r;                                 \
    bb.x = Wi[(size_t)col * LB + (KA)];                \
    bb.y = Wi[(size_t)col * LB + (KA) + 1];            \
    c1 = wmma4(AV, bb, c1);                            \
    col = tn + 32 + r;                                 \
    bb.x = Wi[(size_t)col * LB + (KA)];                \
    bb.y = Wi[(size_t)col * LB + (KA) + 1];            \
    c2 = wmma4(AV, bb, c2);                            \
    col = tn + 48 + r;                                 \
    bb.x = Wi[(size_t)col * LB + (KA)];                \
    bb.y = Wi[(size_t)col * LB + (KA) + 1];            \
    c3 = wmma4(AV, bb, c3);                            \
  }
  // segment 1: memory part, pure vector loads
  for (int k0 = 0; k0 < 128; k0 += 4) {
    int ka = k0 + kh2;
    v2f a = *(const v2f*)(mrow + ka);
    GI_MM(a, ka)
  }
  // segment 2: k0=128 straddles [ef | nf[0..2]] -> cndmask select, no branch
  {
    int ka = 128 + kh2;
    v2f a;
    a.x = half ? frow[1] : efv;
    a.y = half ? frow[2] : frow[0];
    GI_MM(a, ka)
  }
  // segment 3: node-feature part, unconditional scalar loads
  for (int k0 = 132; k0 < 256; k0 += 4) {
    int ka = k0 + kh2;
    v2f a;
    a.x = frow[ka - 129];
    a.y = frow[ka - 128];
    GI_MM(a, ka)
  }
  // segment 4: k0=256 -> only real column 256 (half==0 lanes); pad is zero
  {
    v2f a;
    a.x = half ? 0.f : frow[127];
    a.y = 0.f;
    v2f bb;
    int col = tn + r;
    bb.x = half ? 0.f : Wi[(size_t)col * LB + 256];
    bb.y = 0.f;
    c0 = wmma4(a, bb, c0);
    col = tn + 16 + r;
    bb.x = half ? 0.f : Wi[(size_t)col * LB + 256];
    c1 = wmma4(a, bb, c1);
    col = tn + 32 + r;
    bb.x = half ? 0.f : Wi[(size_t)col * LB + 256];
    c2 = wmma4(a, bb, c2);
    col = tn + 48 + r;
    bb.x = half ? 0.f : Wi[(size_t)col * LB + 256];
    c3 = wmma4(a, bb, c3);
  }
#undef GI_MM
#define GI_STORE(T, CC)                                                      \
  for (int i = 0; i < 8; ++i)                                                \
    gi[(size_t)(tm + i + (half << 3)) * Nt + tn + (T << 4) + r] = CC[i];
  GI_STORE(0, c0)
  GI_STORE(1, c1)
  GI_STORE(2, c2)
  GI_STORE(3, c3)
#undef GI_STORE
}

// ---------------------------------------------------------------------------
// k/v GEMM: rows = N*KNB gathered [h[nbr] (0..127) | efeat (128) |
// cos((pt-t)*w+b) (129..136)], K=137. W stored [137,128] row-major.
// ---------------------------------------------------------------------------
__global__ void k_gemm_kv(const float* __restrict__ h,
                          const int* __restrict__ nbrs,
                          const float* __restrict__ ntimes,
                          const float* __restrict__ nefeat,
                          const float* __restrict__ tw,
                          const float* __restrict__ tb,
                          const int* __restrict__ ptime,
                          const float* __restrict__ W,  // [137,128]
                          float* __restrict__ out) {    // [N*KNB,128]
  const int Nt = 128, M = NN * KNB;
  int wid = blockIdx.x * 8 + (threadIdx.x >> 5);
  int ntiles = Nt >> 6;  // 2
  if (wid >= (M >> 4) * ntiles) return;
  int lane = threadIdx.x & 31;
  int r = lane & 15, half = lane >> 4, kh2 = half << 1;
  int tm = (wid / ntiles) << 4;
  int tn = (wid % ntiles) << 6;
  int row = tm + r;
  int nb = nbrs[row];
  float dt = (float)ptime[0] - ntimes[row];
  float efv = nefeat[row];
  const float* hrow = h + (size_t)nb * DF;
  float kt0 = cosf(dt * tw[0] + tb[0]);
  float kt1 = cosf(dt * tw[1] + tb[1]);
  float kt2 = cosf(dt * tw[2] + tb[2]);
  float kt3 = cosf(dt * tw[3] + tb[3]);
  float kt4 = cosf(dt * tw[4] + tb[4]);
  float kt5 = cosf(dt * tw[5] + tb[5]);
  float kt6 = cosf(dt * tw[6] + tb[6]);
  float kt7 = cosf(dt * tw[7] + tb[7]);
  v8f c0 = {}, c1 = {}, c2 = {}, c3 = {};
#define KV_MM(AV, KA)                              \
  {                                                \
    v2f bb;                                        \
    int col = tn + r;                              \
    bb.x = W[(size_t)(KA)*Nt + col];               \
    bb.y = W[(size_t)((KA) + 1) * Nt + col];       \
    c0 = wmma4(AV, bb, c0);                        \
    col = tn + 16 + r;                             \
    bb.x = W[(size_t)(KA)*Nt + col];               \
    bb.y = W[(size_t)((KA) + 1) * Nt + col];       \
    c1 = wmma4(AV, bb, c1);                        \
    col = tn + 32 + r;                             \
    bb.x = W[(size_t)(KA)*Nt + col];               \
    bb.y = W[(size_t)((KA) + 1) * Nt + col];       \
    c2 = wmma4(AV, bb, c2);                        \
    col = tn + 48 + r;                             \
    bb.x = W[(size_t)(KA)*Nt + col];               \
    bb.y = W[(size_t)((KA) + 1) * Nt + col];       \
    c3 = wmma4(AV, bb, c3);                        \
  }
  for (int k0 = 0; k0 < 128; k0 += 4) {
    int ka = k0 + kh2;
    v2f a = *(const v2f*)(hrow + ka);
    KV_MM(a, ka)
  }
  {  // k0=128: [ef, kt0] / [kt1, kt2]
    v2f a;
    a.x = half ? kt1 : efv;
    a.y = half ? kt2 : kt0;
    KV_MM(a, 128 + kh2)
  }
  {  // k0=132: [kt3, kt4] / [kt5, kt6]
    v2f a;
    a.x = half ? kt5 : kt3;
    a.y = half ? kt6 : kt4;
    KV_MM(a, 132 + kh2)
  }
#undef KV_MM
  {  // k0=136: [kt7, 0] / [0, 0]; guard B row 136 (last real row)
    v2f a;
    a.x = half ? 0.f : kt7;
    a.y = 0.f;
    v2f bb;
    int col = tn + r;
    bb.x = half ? 0.f : W[(size_t)136 * Nt + col];
    bb.y = 0.f;
    c0 = wmma4(a, bb, c0);
    col = tn + 16 + r;
    bb.x = half ? 0.f : W[(size_t)136 * Nt + col];
    c1 = wmma4(a, bb, c1);
    col = tn + 32 + r;
    bb.x = half ? 0.f : W[(size_t)136 * Nt + col];
    c2 = wmma4(a, bb, c2);
    col = tn + 48 + r;
    bb.x = half ? 0.f : W[(size_t)136 * Nt + col];
    c3 = wmma4(a, bb, c3);
  }
#define KV_STORE(T, CC)                                                       \
  for (int i = 0; i < 8; ++i)                                                 \
    out[(size_t)(tm + i + (half << 3)) * Nt + tn + (T << 4) + r] = CC[i];
  KV_STORE(0, c0)
  KV_STORE(1, c1)
  KV_STORE(2, c2)
  KV_STORE(3, c3)
#undef KV_STORE
}

// --------------------------- small helper kernels ---------------------------
__global__ void k_fill_i32(int* p, int v, int n) {
  int i = blockIdx.x * blockDim.x + threadIdx.x;
  if (i < n) p[i] = v;
}
__global__ void k_scan_last(const int* __restrict__ srcC, int* lastpos,
                            int nE) {
  int i = blockIdx.x * blockDim.x + threadIdx.x;
  if (i < nE) atomicMax(&lastpos[srcC[i]], i);
}
__global__ void k_mem_combine(const float* __restrict__ gi,
                              const float* __restrict__ gh,
                              const float* __restrict__ bi,
                              const float* __restrict__ bh,
                              const int* __restrict__ lastpos,
                              const float* __restrict__ tC, float* mem,
                              float* lu) {
  int idx = blockIdx.x * blockDim.x + threadIdx.x;  // N*128
  if (idx >= NN * DF) return;
  int n = idx >> 7, f = idx & 127;
  int lp = lastpos[n];
  if (lp < 0) return;
  size_t g = (size_t)n * 3 * DF;
  float ir = gi[g + f] + bi[f];
  float iz = gi[g + DF + f] + bi[DF + f];
  float ig = gi[g + 2 * DF + f] + bi[2 * DF + f];
  float hr = gh[g + f] + bh[f];
  float hz = gh[g + DF + f] + bh[DF + f];
  float hg = gh[g + 2 * DF + f] + bh[2 * DF + f];
  float rr = sigm(ir + hr), zz = sigm(iz + hz);
  float gg = tanhf(ig + rr * hg);
  float hprev = mem[idx];
  mem[idx] = (1.f - zz) * gg + zz * hprev;
  if (f == 0) lu[n] = tC[lp];
}
__global__ void k_gru_combine(const float* __restrict__ gi,
                              const float* __restrict__ gh,
                              const float* __restrict__ bi,
                              const float* __restrict__ bh,
                              const float* __restrict__ hprev, float* out,
                              int F) {
  int idx = blockIdx.x * blockDim.x + threadIdx.x;  // N*F
  if (idx >= NN * F) return;
  int n = idx / F, f = idx % F;
  size_t g = (size_t)n * 3 * F;
  float ir = gi[g + f] + bi[f];
  float iz = gi[g + F + f] + bi[F + f];
  float ig = gi[g + 2 * F + f] + bi[2 * F + f];
  float hr = gh[g + f] + bh[f];
  float hz = gh[g + F + f] + bh[F + f];
  float hg = gh[g + 2 * F + f] + bh[2 * F + f];
  float rr = sigm(ir + hr), zz = sigm(iz + hz);
  float gg = tanhf(ig + rr * hg);
  out[idx] = (1.f - zz) * gg + zz * hprev[idx];
}
__global__ void k_add(const float* a, const float* b, float* o, int n) {
  int i = blockIdx.x * blockDim.x + threadIdx.x;
  if (i < n) o[i] = a[i] + b[i];
}
__global__ void k_build_qin(const float* __restrict__ h,
                            const float* __restrict__ tb, float* qin) {
  int idx = blockIdx.x * blockDim.x + threadIdx.x;  // N*136
  if (idx >= NN * 136) return;
  int n = idx / 136, k = idx % 136;
  qin[idx] = (k < 128) ? h[(size_t)n * DF + k] : cosf(tb[k - 128]);
}
__global__ void k_build_merged(const float* __restrict__ attnb,
                               const float* __restrict__ nf, float* m) {
  int idx = blockIdx.x * blockDim.x + threadIdx.x;  // N*256
  if (idx >= NN * 256) return;
  int n = idx >> 8, k = idx & 255;
  m[idx] = (k < 128) ? attnb[(size_t)n * DF + k] : nf[(size_t)n * DF + k - 128];
}
__global__ void k_build_cat2(const float* __restrict__ emb,
                             const float* __restrict__ dtdg, float* c) {
  int idx = blockIdx.x * blockDim.x + threadIdx.x;  // N*640
  if (idx >= NN * 640) return;
  int n = idx / 640, k = idx % 640;
  c[idx] = (k < 128) ? emb[(size_t)n * DF + k] : dtdg[(size_t)n * HD + k - 128];
}
__global__ void k_emb_gather(const float* __restrict__ h,
                             const int* __restrict__ pred, float* emb) {
  int idx = blockIdx.x * blockDim.x + threadIdx.x;  // N*128
  if (idx >= NN * DF) return;
  int n = idx >> 7, f = idx & 127;
  emb[idx] = h[(size_t)pred[n] * DF + f];
}
__global__ void k_copy_f32(const float* s, float* d, int n) {
  int i = blockIdx.x * blockDim.x + threadIdx.x;
  if (i < n) d[i] = s[i];
}

// 20-neighbor, 2-head softmax attention; one wave per (node, head)
__global__ void k_attn(const float* __restrict__ q,
                       const float* __restrict__ kmat,
                       const float* __restrict__ vmat, float* __restrict__ o) {
  int wid = blockIdx.x * 8 + (threadIdx.x >> 5);  // NN*2 waves
  if (wid >= NN * 2) return;
  int lane = threadIdx.x & 31;
  int n = wid >> 1, hh = wid & 1, base = hh * 64;
  const float* qrow = q + (size_t)n * DF + base;
  float s = -1e30f;
  if (lane < KNB) {
    const float* krow = kmat + (size_t)(n * KNB + lane) * DF + base;
    float acc = 0.f;
    for (int c = 0; c < 64; ++c) acc += qrow[c] * krow[c];
    s = acc * 0.125f;  // 1/sqrt(64)
  }
  float m = s;
  for (int off = 16; off; off >>= 1) m = fmaxf(m, __shfl_xor(m, off));
  float e = (lane < KNB) ? expf(s - m) : 0.f;
  float sum = e;
  for (int off = 16; off; off >>= 1) sum += __shfl_xor(sum, off);
  float w = e / sum;
  for (int cc = lane; cc < 64; cc += 32) {
    float acc = 0.f;
    for (int j = 0; j < KNB; ++j) {
      float wj = __shfl(w, j);
      acc += wj * vmat[(size_t)(n * KNB + j) * DF + base + cc];
    }
    o[(size_t)n * DF + base + cc] = acc;
  }
}

static inline int gb32(int M, int Nt) { return ((M / 32) * (Nt / 64) + 7) / 8; }

extern "C" void kernel_launch(void* const* d_in, const int* in_sizes, int n_in,
                              void* d_out, int out_size, void* d_ws,
                              size_t ws_size, hipStream_t stream) {
  (void)in_sizes; (void)n_in; (void)out_size; (void)ws_size;
  const int* src = (const int*)d_in[0];
  const int* dst = (const int*)d_in[1];
  const float* etimes = (const float*)d_in[2];
  const float* efeat = (const float*)d_in[3];
  const float* input_od = (const float*)d_in[4];
  const int* pred = (const int*)d_in[5];
  const int* ptime = (const int*)d_in[6];
  const int* nbrs = (const int*)d_in[7];
  const float* ntimes = (const float*)d_in[8];
  const float* nefeat = (const float*)d_in[9];
  const float* nf = (const float*)d_in[10];
  const float* memory0 = (const float*)d_in[11];
  const float* lu0 = (const float*)d_in[12];
  const float* tw = (const float*)d_in[13];
  const float* tb = (const float*)d_in[14];
  const float* Wi_m = (const float*)d_in[15];
  const float* Wh_m = (const float*)d_in[16];
  const float* bi_m = (const float*)d_in[17];
  const float* bh_m = (const float*)d_in[18];
  const float* Wi0 = (const float*)d_in[19];
  const float* Wh0 = (const float*)d_in[20];
  const float* bi0 = (const float*)d_in[21];
  const float* bh0 = (const float*)d_in[22];
  const float* Wi1 = (const float*)d_in[23];
  const float* Wh1 = (const float*)d_in[24];
  const float* bi1 = (const float*)d_in[25];
  const float* bh1 = (const float*)d_in[26];
  const float* h0 = (const float*)d_in[27];  // [2,N,H]
  const float* Wq = (const float*)d_in[28];  // [2,136,128]
  const float* Wk = (const float*)d_in[29];  // [2,137,128]
  const float* Wv = (const float*)d_in[30];  // [2,137,128]
  const float* fc1w = (const float*)d_in[31];
  const float* fc1b = (const float*)d_in[32];
  const float* fc2w = (const float*)d_in[33];
  const float* fc2b = (const float*)d_in[34];
  const float* W1 = (const float*)d_in[35];
  const float* b1 = (const float*)d_in[36];
  const float* W2 = (const float*)d_in[37];
  const float* b2 = (const float*)d_in[38];

  // ---- workspace layout (floats) ----
  float* W = (float*)d_ws;
  size_t off = 0;
  float* mem = W + off; off += (size_t)NN * DF;
  float* lu = W + off; off += NN;
  int* lastpos = (int*)(W + off); off += NN;
  float* dtdg = W + off; off += (size_t)NN * HD;
  float* h = W + off; off += (size_t)NN * DF;
  float* attnb = W + off; off += (size_t)NN * DF;
  float* qin = W + off; off += (size_t)NN * 136;
  float* qmat = W + off; off += (size_t)NN * DF;
  float* merged = W + off; off += (size_t)NN * 256;
  float* htmp = W + off; off += (size_t)NN * DF;
  float* cat2 = W + off; off += (size_t)NN * 640;
  float* hid = W + off; off += (size_t)NN * DF;
  float* S = W + off;  // big union scratch: max 40960*128*2 floats
  float* gi = S;                       // scan phase [N,384]
  float* gh = S + (size_t)NN * 384;
  float* gi0 = S;                      // dtdg phase [N,1536]
  float* gh0 = S + (size_t)NN * 1536;
  float* hl0 = S + (size_t)NN * 3072;  // [N,512]
  float* kmat = S;                     // attn phase [N*20,128]
  float* vmat = S + (size_t)NN * KNB * DF;

  float* out_od = (float*)d_out;              // [N,N]
  float* out_emb = out_od + (size_t)NN * NN;  // [N,128]
  float* out_lu = out_emb + (size_t)NN * DF;  // [N]

  hipMemcpyAsync(mem, memory0, sizeof(float) * NN * DF,
                 hipMemcpyDeviceToDevice, stream);
  hipMemcpyAsync(lu, lu0, sizeof(float) * NN, hipMemcpyDeviceToDevice, stream);

  // ---- 1) sequential memory scan: 16 chunks of last-aggregation + GRU ----
  for (int c = 0; c < NCHUNK; ++c) {
    const int* srcC = src + c * DIVB;
    const int* dstC = dst + c * DIVB;
    const float* efC = efeat + c * DIVB;
    const float* tC = etimes + c * DIVB;
    k_fill_i32<<<NN / 256, 256, 0, stream>>>(lastpos, -1, NN);
    k_scan_last<<<DIVB / 256, 256, 0, stream>>>(srcC, lastpos, DIVB);
    k_gemm_gi_mem<<<((NN / 16) * 6 + 7) / 8, 256, 0, stream>>>(
        mem, nf, lastpos, dstC, efC, Wi_m, gi);
    k_gemm<true><<<gb32(NN, 384), 256, 0, stream>>>(
        mem, DF, Wh_m, DF, nullptr, gh, 384, NN, 384, DF, 0);
    k_mem_combine<<<NN * DF / 256, 256, 0, stream>>>(gi, gh, bi_m, bh_m,
                                                     lastpos, tC, mem, lu);
  }

  // ---- 2) DTDG 2-layer GRU ----
  k_gemm<true><<<gb32(NN, 1536), 256, 0, stream>>>(
      input_od, NN, Wi0, NN, nullptr, gi0, 1536, NN, 1536, NN, 0);
  k_gemm<true><<<gb32(NN, 1536), 256, 0, stream>>>(
      h0, HD, Wh0, HD, nullptr, gh0, 1536, NN, 1536, HD, 0);
  k_gru_combine<<<NN * HD / 256, 256, 0, stream>>>(gi0, gh0, bi0, bh0, h0, hl0,
                                                   HD);
  k_gemm<true><<<gb32(NN, 1536), 256, 0, stream>>>(
      hl0, HD, Wi1, HD, nullptr, gi0, 1536, NN, 1536, HD, 0);
  k_gemm<true><<<gb32(NN, 1536), 256, 0, stream>>>(
      h0 + (size_t)NN * HD, HD, Wh1, HD, nullptr, gh0, 1536, NN, 1536, HD, 0);
  k_gru_combine<<<NN * HD / 256, 256, 0, stream>>>(
      gi0, gh0, bi1, bh1, h0 + (size_t)NN * HD, dtdg, HD);

  // ---- 3) two layers of temporal graph attention ----
  k_add<<<NN * DF / 256, 256, 0, stream>>>(mem, nf, h, NN * DF);
  for (int l = 0; l < 2; ++l) {
    k_build_qin<<<NN * 136 / 256, 256, 0, stream>>>(h, tb, qin);
    k_gemm<false><<<gb32(NN, DF), 256, 0, stream>>>(
        qin, 136, Wq + (size_t)l * 136 * DF, DF, nullptr, qmat, DF, NN, DF,
        136, 0);
    k_gemm_kv<<<((NN * KNB / 16) * 2 + 7) / 8, 256, 0, stream>>>(
        h, nbrs, ntimes, nefeat, tw, tb, ptime, Wk + (size_t)l * 137 * DF,
        kmat);
    k_gemm_kv<<<((NN * KNB / 16) * 2 + 7) / 8, 256, 0, stream>>>(
        h, nbrs, ntimes, nefeat, tw, tb, ptime, Wv + (size_t)l * 137 * DF,
        vmat);
    k_attn<<<(NN * 2 + 7) / 8, 256, 0, stream>>>(qmat, kmat, vmat, attnb);
    k_build_merged<<<NN * 256 / 256, 256, 0, stream>>>(attnb, nf, merged);
    k_gemm<false><<<gb32(NN, DF), 256, 0, stream>>>(
        merged, 256, fc1w + (size_t)l * 256 * DF, DF, fc1b + l * DF, htmp, DF,
        NN, DF, 256, 1);
    k_gemm<false><<<gb32(NN, DF), 256, 0, stream>>>(
        htmp, DF, fc2w + (size_t)l * DF * DF, DF, fc2b + l * DF, h, DF, NN, DF,
        DF, 0);
  }

  // ---- 4) outputs: emb gather, OD head, last_update ----
  k_emb_gather<<<NN * DF / 256, 256, 0, stream>>>(h, pred, out_emb);
  k_build_cat2<<<NN * 640 / 256, 256, 0, stream>>>(out_emb, dtdg, cat2);
  k_gemm<false><<<gb32(NN, DF), 256, 0, stream>>>(cat2, 640, W1, DF, b1, hid,
                                                  DF, NN, DF, 640, 1);
  k_gemm<false><<<gb32(NN, NN), 256, 0, stream>>>(hid, DF, W2, NN, b2, out_od,
                                                  NN, NN, NN, DF, 0);
  k_copy_f32<<<NN / 256, 256, 0, stream>>>(lu, out_lu, NN);
}